// Embedder_7284264534032
// MI455X (gfx1250) — compile-verified
//
#include <hip/hip_runtime.h>

// Embedding gather + bias + positional add for MI455X (gfx1250).
//
// out[b,s,:] = W_T[text[b,s],:] + bias[:] + pe[s,:]   B=4, S=2048, D=1024, fp32
//
// Pure bandwidth problem (~72 MB of traffic -> ~3.1 us at 23.3 TB/s). No
// matmul, so no WMMA. The CDNA5 feature that fits is the async global->LDS
// DMA path (ASYNCcnt): the gathered 4 KB W_T row is DMA'd into LDS with
// global_load_async_to_lds_b128 while the sequential pe/bias loads are in
// flight, then the wave waits on s_wait_asynccnt 0 and finishes the adds.
// All accesses are 16 B-aligned b128 ops, coalesced per wave32.

#define D_MODEL 1024
#define SEQ     2048

#define __AS1 __attribute__((address_space(1)))
#define __AS3 __attribute__((address_space(3)))

typedef float f32x4 __attribute__((ext_vector_type(4)));
typedef int   v4i   __attribute__((ext_vector_type(4)));

__device__ __forceinline__ void async_row_b128(const float* gsrc, float* lds_dst) {
#if __has_builtin(__builtin_amdgcn_global_load_async_to_lds_b128)
    __builtin_amdgcn_global_load_async_to_lds_b128(
        (__AS1 v4i*)gsrc, (__AS3 v4i*)lds_dst, /*offset=*/0, /*cpol=*/0);
#else
    unsigned long long gaddr = (unsigned long long)gsrc;
    unsigned lds_off = (unsigned)(unsigned long long)(__AS3 void*)lds_dst;
    asm volatile("global_load_async_to_lds_b128 %0, %1, off"
                 :: "v"(lds_off), "v"(gaddr)
                 : "memory");
#endif
}

__device__ __forceinline__ void wait_async0() {
#if __has_builtin(__builtin_amdgcn_s_wait_asynccnt)
    __builtin_amdgcn_s_wait_asynccnt(0);
#else
    asm volatile("s_wait_asynccnt 0" ::: "memory");
#endif
}

__global__ __launch_bounds__(256)
void embed_bias_pe_kernel(const int*   __restrict__ text,   // [4,2048]
                          const float* __restrict__ W_T,    // [50257,1024]
                          const float* __restrict__ bias,   // [1024]
                          const float* __restrict__ pe,     // [1,4096,1024]
                          float*       __restrict__ out)    // [4,2048,1024]
{
    __shared__ float row[D_MODEL];               // 4 KB staging for gathered row

    const int token = blockIdx.x;                // flat index over [B,S]
    const int s     = token & (SEQ - 1);         // sequence position
    const int d     = threadIdx.x * 4;           // this thread's float4 slot

    // Uniform per block -> scalar load of the token id.
    const int tok = text[token];

    // Kick off the gathered-row DMA into LDS (ASYNCcnt path). Each lane moves
    // its own 16 B; a wave32 covers a contiguous 512 B of the 4 KB row.
    const float* src = W_T + (size_t)tok * D_MODEL + d;
    async_row_b128(src, &row[d]);

    // Overlap: sequential pe row + bias (L2-friendly) while the DMA flies.
    f32x4 pv = *(const f32x4*)(pe  + (size_t)s * D_MODEL + d);
    f32x4 bv = *(const f32x4*)(bias + d);

    // Wave-local wait: each lane reads back exactly the bytes it DMA'd, so
    // ASYNCcnt==0 is sufficient (no cross-lane sharing, no barrier needed).
    wait_async0();

    f32x4 ev = *(const f32x4*)(&row[d]);
    f32x4 ov = ev + bv + pv;
    *(f32x4*)(out + (size_t)token * D_MODEL + d) = ov;
}

extern "C" void kernel_launch(void* const* d_in, const int* in_sizes, int n_in,
                              void* d_out, int out_size, void* d_ws, size_t ws_size,
                              hipStream_t stream) {
    const int*   text = (const int*)  d_in[0];   // [4,2048] int32
    const float* W_T  = (const float*)d_in[1];   // [50257,1024] f32
    const float* bias = (const float*)d_in[2];   // [1024] f32
    const float* pe   = (const float*)d_in[3];   // [1,4096,1024] f32

    float* out = (float*)d_out;                  // [4,2048,1024] f32

    const int n_tokens = in_sizes[0];            // 8192
    dim3 grid(n_tokens);
    dim3 block(256);                             // 8 wave32s, one float4/thread
    hipLaunchKernelGGL(embed_bias_pe_kernel, grid, block, 0, stream,
                       text, W_T, bias, pe, out);
}